// DirectOrderingModel_22179211117310
// MI455X (gfx1250) — compile-verified
//
#include <hip/hip_runtime.h>
#include <hip/hip_bf16.h>
#include <math.h>

typedef __attribute__((ext_vector_type(2))) float v2f;
typedef __attribute__((ext_vector_type(8))) float v8f;

#define B_ROWS 4096
#define H_DIM  1024
#define D_DIM  4

__device__ __forceinline__ float gelu_exact(float x) {
  // exact GELU: 0.5*x*(1+erf(x/sqrt(2))) — matches torch nn.GELU / jax approximate=False
  return 0.5f * x * (1.0f + erff(x * 0.70710678118654752440f));
}

// ---------------------------------------------------------------------------
// accumulator init: acc[0]=sse, acc[1]=bce_sum (doubles), *cnt = masked count
// ---------------------------------------------------------------------------
__global__ void init_acc_kernel(double* acc, unsigned* cnt) {
  if (threadIdx.x == 0) { acc[0] = 0.0; acc[1] = 0.0; *cnt = 0u; }
}

// ---------------------------------------------------------------------------
// row-wise LayerNorm: one block (256 threads) per row
// ---------------------------------------------------------------------------
__global__ void __launch_bounds__(256)
ln_kernel(const float* __restrict__ x, const float* __restrict__ g,
          const float* __restrict__ bta, float* __restrict__ y, int H) {
  __shared__ float red[256];
  const int row = blockIdx.x;
  const int t = threadIdx.x;
  const float* xr = x + (size_t)row * H;
  float* yr = y + (size_t)row * H;

  float s = 0.f;
  for (int i = t; i < H; i += 256) s += xr[i];
  red[t] = s; __syncthreads();
  for (int off = 128; off > 0; off >>= 1) {
    if (t < off) red[t] += red[t + off];
    __syncthreads();
  }
  const float mu = red[0] * (1.0f / (float)H);
  __syncthreads();

  float v = 0.f;
  for (int i = t; i < H; i += 256) { float d = xr[i] - mu; v += d * d; }
  red[t] = v; __syncthreads();
  for (int off = 128; off > 0; off >>= 1) {
    if (t < off) red[t] += red[t + off];
    __syncthreads();
  }
  const float rstd = rsqrtf(red[0] * (1.0f / (float)H) + 1e-5f);

  for (int i = t; i < H; i += 256)
    yr[i] = (xr[i] - mu) * rstd * g[i] + bta[i];
}

// ---------------------------------------------------------------------------
// fp32 WMMA GEMM + bias + exact GELU.  C[M,N] = gelu(A[M,K] @ B[K,N] + bias)
// One wave32 per 16x16 output tile; V_WMMA_F32_16X16X4_F32 over K in steps of 4.
// Fragment layouts per CDNA5 ISA §7.12.2:
//   A 16x4 : lane m=lane&15 holds K = 2*(lane>>4)+{0,1}  (contiguous float2)
//   B 4x16 : lane n=lane&15 holds rows K = 2*(lane>>4)+{0,1}
//   C 16x16: vgpr r -> row r + 8*(lane>>4), col lane&15
// ---------------------------------------------------------------------------
__global__ void __launch_bounds__(256)
wmma_gemm_bias_gelu(const float* __restrict__ A, const float* __restrict__ Bm,
                    const float* __restrict__ bias, float* __restrict__ C,
                    int M, int N, int K) {
  const int lane  = threadIdx.x & 31;
  const int wave  = threadIdx.x >> 5;
  const int tile  = blockIdx.x * (blockDim.x >> 5) + wave;
  const int tilesN = N >> 4;
  const int tm = tile / tilesN;
  const int tn = tile - tm * tilesN;
  if (tm * 16 >= M) return;           // wave-uniform guard (EXEC all-1 inside)

  const int r16   = lane & 15;
  const int khalf = lane >> 4;        // 0 or 1
  const int m = tm * 16 + r16;        // A row this lane loads
  const int n = tn * 16 + r16;        // B/C column this lane owns

  const float* Arow = A  + (size_t)m * K + 2 * khalf;
  const float* Bcol = Bm + (size_t)(2 * khalf) * N + n;

  v8f c = {};
  for (int k = 0; k < K; k += 4) {
    v2f a, b;
    a.x = Arow[k];
    a.y = Arow[k + 1];
    b.x = Bcol[(size_t)k * N];
    b.y = Bcol[(size_t)k * N + N];
    c = __builtin_amdgcn_wmma_f32_16x16x4_f32(
        /*neg_a=*/false, a, /*neg_b=*/false, b,
        /*c_mod=*/(short)0, c, /*reuse_a=*/false, /*reuse_b=*/false);
  }

  const float bn = bias[n];
#pragma unroll
  for (int r = 0; r < 8; ++r) {
    const int row = tm * 16 + khalf * 8 + r;
    C[(size_t)row * N + n] = gelu_exact(c[r] + bn);
  }
}

// ---------------------------------------------------------------------------
// head: logits = sigmoid(h2 @ Wh + bh), one thread per (row, d) element
// ---------------------------------------------------------------------------
__global__ void __launch_bounds__(256)
head_kernel(const float* __restrict__ h2, const float* __restrict__ Wh,
            const float* __restrict__ bh, float* __restrict__ logits) {
  const int idx = blockIdx.x * blockDim.x + threadIdx.x;
  if (idx >= B_ROWS * D_DIM) return;
  const int row = idx >> 2;
  const int d   = idx & 3;
  const float* hr = h2 + (size_t)row * (H_DIM / 4);
  float acc = bh[d];
  for (int k = 0; k < H_DIM / 4; ++k)
    acc = fmaf(hr[k], Wh[k * D_DIM + d], acc);
  logits[idx] = 1.0f / (1.0f + expf(-acc));
}

// ---------------------------------------------------------------------------
// MSE sum-of-squares reduction into acc[0]
// ---------------------------------------------------------------------------
__global__ void __launch_bounds__(256)
mse_kernel(const float* __restrict__ logits, const float* __restrict__ labels,
           double* __restrict__ acc) {
  __shared__ double red[256];
  const int idx = blockIdx.x * blockDim.x + threadIdx.x;
  float d = 0.f;
  if (idx < B_ROWS * D_DIM) d = logits[idx] - labels[idx];
  red[threadIdx.x] = (double)d * (double)d;
  __syncthreads();
  for (int off = 128; off > 0; off >>= 1) {
    if (threadIdx.x < off) red[threadIdx.x] += red[threadIdx.x + off];
    __syncthreads();
  }
  if (threadIdx.x == 0) atomicAdd(&acc[0], red[0]);
}

// ---------------------------------------------------------------------------
// pairwise ordering loss: blocks tile (i,j); only upper triangle does work.
// bce sum -> acc[1] (double atomics), exact mask count -> *cnt (uint atomics)
// ---------------------------------------------------------------------------
__global__ void __launch_bounds__(256)
pair_kernel(const float* __restrict__ logits, const float* __restrict__ labels,
            double* __restrict__ acc, unsigned* __restrict__ cnt) {
  const int bx = blockIdx.x;   // j tile
  const int by = blockIdx.y;   // i tile
  if (by > bx) return;         // tile entirely below diagonal

  __shared__ double   reds[256];
  __shared__ unsigned redc[256];

  const int ti = threadIdx.x >> 4;
  const int tj = threadIdx.x & 15;
  const int i = by * 16 + ti;
  const int j = bx * 16 + tj;

  float    lsum = 0.f;
  unsigned lcnt = 0u;
  if (i < j) {
#pragma unroll
    for (int d = 0; d < D_DIM; ++d) {
      const float ld = labels[i * D_DIM + d] - labels[j * D_DIM + d];
      if (fabsf(ld) > 0.01f) {
        const float pd = logits[i * D_DIM + d] - logits[j * D_DIM + d];
        const float tp = 1.0f / (1.0f + expf(-20.0f * ld));
        const float pp = 1.0f / (1.0f + expf(-20.0f * pd));
        const float bce = -tp * logf(pp + 1e-7f)
                          - (1.0f - tp) * logf(1.0f - pp + 1e-7f);
        lsum += bce;
        lcnt += 1u;
      }
    }
  }
  reds[threadIdx.x] = (double)lsum;
  redc[threadIdx.x] = lcnt;
  __syncthreads();
  for (int off = 128; off > 0; off >>= 1) {
    if (threadIdx.x < off) {
      reds[threadIdx.x] += reds[threadIdx.x + off];
      redc[threadIdx.x] += redc[threadIdx.x + off];
    }
    __syncthreads();
  }
  if (threadIdx.x == 0 && (redc[0] != 0u)) {
    atomicAdd(&acc[1], reds[0]);
    atomicAdd(cnt, redc[0]);
  }
}

// ---------------------------------------------------------------------------
// finalize: loss = 0.5*mse + 0.5*ordering  ->  d_out[0]
// ---------------------------------------------------------------------------
__global__ void finalize_kernel(const double* __restrict__ acc,
                                const unsigned* __restrict__ cnt,
                                float* __restrict__ out) {
  if (threadIdx.x == 0) {
    const double mse = acc[0] / (double)(B_ROWS * D_DIM);
    const unsigned c = *cnt;
    const double ord = (c > 0u) ? (acc[1] / (double)c) : 0.0;
    out[0] = (float)(0.5 * mse + 0.5 * ord);
  }
}

// ---------------------------------------------------------------------------
extern "C" void kernel_launch(void* const* d_in, const int* in_sizes, int n_in,
                              void* d_out, int out_size, void* d_ws, size_t ws_size,
                              hipStream_t stream) {
  const float* pooled = (const float*)d_in[0];
  const float* labels = (const float*)d_in[1];
  const float* ln1_g  = (const float*)d_in[2];
  const float* ln1_b  = (const float*)d_in[3];
  const float* W1     = (const float*)d_in[4];
  const float* b1     = (const float*)d_in[5];
  const float* ln2_g  = (const float*)d_in[6];
  const float* ln2_b  = (const float*)d_in[7];
  const float* W2     = (const float*)d_in[8];
  const float* b2     = (const float*)d_in[9];
  const float* Wh     = (const float*)d_in[10];
  const float* bh     = (const float*)d_in[11];
  (void)in_sizes; (void)n_in; (void)out_size; (void)ws_size;

  float* out    = (float*)d_out;
  float* logits = out + 1;                 // d_out = [loss, logits(4096*4)]

  char* ws = (char*)d_ws;
  double*   acc  = (double*)ws;            // 2 doubles at offset 0 (aligned)
  unsigned* cnt  = (unsigned*)(ws + 16);
  float* xhat1 = (float*)(ws + 64);
  float* h1    = xhat1 + (size_t)B_ROWS * H_DIM;
  float* xhat2 = h1    + (size_t)B_ROWS * (H_DIM / 2);
  float* h2    = xhat2 + (size_t)B_ROWS * (H_DIM / 2);

  init_acc_kernel<<<1, 32, 0, stream>>>(acc, cnt);

  // xhat1 = LN(pooled)
  ln_kernel<<<B_ROWS, 256, 0, stream>>>(pooled, ln1_g, ln1_b, xhat1, H_DIM);

  // h1 = gelu(xhat1 @ W1 + b1)   [4096x1024]x[1024x512]
  {
    const int M = B_ROWS, N = H_DIM / 2, K = H_DIM;
    const int tiles = (M / 16) * (N / 16);       // 8192 waves
    wmma_gemm_bias_gelu<<<tiles / 8, 256, 0, stream>>>(xhat1, W1, b1, h1, M, N, K);
  }

  // xhat2 = LN(h1)
  ln_kernel<<<B_ROWS, 256, 0, stream>>>(h1, ln2_g, ln2_b, xhat2, H_DIM / 2);

  // h2 = gelu(xhat2 @ W2 + b2)   [4096x512]x[512x256]
  {
    const int M = B_ROWS, N = H_DIM / 4, K = H_DIM / 2;
    const int tiles = (M / 16) * (N / 16);       // 4096 waves
    wmma_gemm_bias_gelu<<<tiles / 8, 256, 0, stream>>>(xhat2, W2, b2, h2, M, N, K);
  }

  // logits = sigmoid(h2 @ Wh + bh) -> straight into d_out+1
  head_kernel<<<(B_ROWS * D_DIM) / 256, 256, 0, stream>>>(h2, Wh, bh, logits);

  // loss pieces
  mse_kernel<<<(B_ROWS * D_DIM + 255) / 256, 256, 0, stream>>>(logits, labels, acc);
  dim3 pg(B_ROWS / 16, B_ROWS / 16);
  pair_kernel<<<pg, 256, 0, stream>>>(logits, labels, acc, cnt);

  finalize_kernel<<<1, 32, 0, stream>>>(acc, cnt, out);
}